// NormalizingFlow_old_21638045237548
// MI455X (gfx1250) — compile-verified
//
#include <hip/hip_runtime.h>
#include <hip/hip_bf16.h>

typedef __attribute__((ext_vector_type(2))) float v2f;
typedef __attribute__((ext_vector_type(8))) float v8f;

#define D 128
#define L 8
#define FLOW_EPS 1e-8f

// Workspace layout (floats)
#define WS_WPACK 0      // 1024 x v2f  (B-operand layout of W^T, K=128 in 32 chunks)
#define WS_UPACK 2048   // 512  x v2f  (B-operand layout of U, 2 K-chunks x 8 col-tiles)
#define WS_G     3072   // G[j][l] = dot(u_j, w_l), 64 floats
#define WS_B     3136   // b[l], 8 floats

// ---------------------------------------------------------------------------
// Setup: precompute G = U.W^T (8x8), and pack W / U into the per-lane VGPR
// layouts required by V_WMMA_F32_16X16X4_F32's B operand:
//   lane ln (n = ln&15, h = ln>>4) holds { B[2h][n], B[2h+1][n] }.
// ---------------------------------------------------------------------------
__global__ void planar_setup_kernel(const float* __restrict__ u,
                                    const float* __restrict__ w,
                                    const float* __restrict__ b,
                                    float* __restrict__ ws) {
  const int tid = threadIdx.x;  // 256 threads, 1 block
  if (tid < 64) {
    const int j = tid >> 3, l = tid & 7;
    float s = 0.f;
    for (int d = 0; d < D; ++d) s += u[j * D + d] * w[l * D + d];
    ws[WS_G + tid] = s;
  }
  if (tid < L) ws[WS_B + tid] = b[tid];

  // Wpack[k][lane]: B chunk for K-step k of S = Z.W^T  (cols l>=8 zero-padded)
  v2f* Wp = (v2f*)(ws + WS_WPACK);
  for (int idx = tid; idx < 32 * 32; idx += 256) {
    const int k = idx >> 5, ln = idx & 31;
    const int n = ln & 15, h = ln >> 4;
    const int kk = 4 * k + 2 * h;
    v2f v;
    v.x = (n < L) ? w[n * D + kk]     : 0.f;
    v.y = (n < L) ? w[n * D + kk + 1] : 0.f;
    Wp[idx] = v;
  }
  // Upack[c][ct][lane]: B chunk for zf = Z0 + T.U ; c = K-chunk (layers 4c..4c+3),
  // ct = 16-col tile of D.
  v2f* Up = (v2f*)(ws + WS_UPACK);
  for (int idx = tid; idx < 2 * 8 * 32; idx += 256) {
    const int c = idx >> 8, ct = (idx >> 5) & 7, ln = idx & 31;
    const int n = ln & 15, h = ln >> 4;
    const int r = 4 * c + 2 * h;
    v2f v;
    v.x = u[r * D + 16 * ct + n];
    v.y = u[(r + 1) * D + 16 * ct + n];
    Up[idx] = v;
  }
}

// ---------------------------------------------------------------------------
// Main kernel: one wave per 16-row tile, 4 waves (128 threads) per block.
// Phase 1: S = Z.W^T via 32x v_wmma_f32_16x16x4_f32 (K=128).
// Phase 2: per-row 8-step tanh recurrence (via per-wave LDS transpose) + ldj.
// Phase 3: zf = Z0 + T.U via 2x WMMA per 16x16 col-tile (z0 re-read hits L2).
// ---------------------------------------------------------------------------
__global__ __launch_bounds__(128) void planar_flow_kernel(
    const float* __restrict__ z, const float* __restrict__ ws,
    float* __restrict__ zf, float* __restrict__ ldj_out) {
  __shared__ float st[4][16][16];  // S tile per wave (WMMA C layout dump)
  __shared__ float tt[4][16][8];   // t values per wave

  const int tid = threadIdx.x;
  const int wv  = tid >> 5;
  const int ln  = tid & 31;
  const int m   = ln & 15;    // row within tile (A/C row, also C col index n)
  const int h   = ln >> 4;    // lane-half selector
  const int n   = m;          // C/B column index for this lane
  const long row0 = ((long)blockIdx.x * 4 + wv) * 16;

  const v2f*  Wp = (const v2f*)(ws + WS_WPACK);
  const v2f*  Up = (const v2f*)(ws + WS_UPACK);
  const float* G  = ws + WS_G;
  const float* bb = ws + WS_B;

  // ---- Phase 1: S accumulation over K=128 -------------------------------
  v8f acc = {0.f, 0.f, 0.f, 0.f, 0.f, 0.f, 0.f, 0.f};
  const float* zr = z + (size_t)(row0 + m) * D + 2 * h;
#pragma unroll 8
  for (int k = 0; k < 32; ++k) {
    v2f av = *(const v2f*)(zr + 4 * k);        // A[m][4k+2h], A[m][4k+2h+1]
    v2f bv = Wp[k * 32 + ln];                  // prepacked B chunk
    acc = __builtin_amdgcn_wmma_f32_16x16x4_f32(false, av, false, bv,
                                                (short)0, acc, false, false);
  }
  // C layout: VGPR i, lane ln -> S[row = i + 8h][layer = n]
#pragma unroll
  for (int i = 0; i < 8; ++i) st[wv][i + 8 * h][n] = acc[i];
  asm volatile("s_wait_dscnt 0" ::: "memory");

  // ---- Phase 2: layer-serial recurrence per row -------------------------
  // z_l.w_l = S_l + sum_{j<l} G[j][l] * t_j   (G[j][l] = u_j.w_l)
  float t[L];
  float ldj = 0.f;
#pragma unroll
  for (int l = 0; l < L; ++l) {
    float a = st[wv][m][l] + bb[l];
#pragma unroll
    for (int j = 0; j < l; ++j) a += G[j * 8 + l] * t[j];
    t[l] = tanhf(a);
    const float det = fabsf(1.f + (1.f - t[l] * t[l]) * G[l * 8 + l]);
    ldj += logf(det + FLOW_EPS);
    tt[wv][m][l] = t[l];
  }
  if (ln < 16) ldj_out[row0 + m] = ldj;
  asm volatile("s_wait_dscnt 0" ::: "memory");

  // ---- Phase 3: zf = Z0 + T.U (K=8 -> 2 WMMA per 16-col tile) -----------
  v2f a0, a1;  // A operand: t[m][K-chunk] in WMMA A layout
  a0.x = tt[wv][m][2 * h];     a0.y = tt[wv][m][2 * h + 1];
  a1.x = tt[wv][m][4 + 2 * h]; a1.y = tt[wv][m][4 + 2 * h + 1];
#pragma unroll
  for (int ct = 0; ct < 8; ++ct) {
    v8f cc;
#pragma unroll
    for (int i = 0; i < 8; ++i)
      cc[i] = z[(size_t)(row0 + i + 8 * h) * D + 16 * ct + n];  // L2 hit
    v2f b0 = Up[ct * 32 + ln];
    v2f b1 = Up[(8 + ct) * 32 + ln];
    cc = __builtin_amdgcn_wmma_f32_16x16x4_f32(false, a0, false, b0,
                                               (short)0, cc, false, false);
    cc = __builtin_amdgcn_wmma_f32_16x16x4_f32(false, a1, false, b1,
                                               (short)0, cc, false, false);
#pragma unroll
    for (int i = 0; i < 8; ++i)
      zf[(size_t)(row0 + i + 8 * h) * D + 16 * ct + n] = cc[i];
  }
}

// ---------------------------------------------------------------------------
extern "C" void kernel_launch(void* const* d_in, const int* in_sizes, int n_in,
                              void* d_out, int out_size, void* d_ws,
                              size_t ws_size, hipStream_t stream) {
  const float* z = (const float*)d_in[0];
  const float* u = (const float*)d_in[1];
  const float* w = (const float*)d_in[2];
  const float* b = (const float*)d_in[3];
  float* ws = (float*)d_ws;

  const long nrows = (long)in_sizes[0] / D;  // P*B = 524288
  float* zf  = (float*)d_out;
  float* ldj = (float*)d_out + (size_t)nrows * D;

  hipLaunchKernelGGL(planar_setup_kernel, dim3(1), dim3(256), 0, stream,
                     u, w, b, ws);
  const long rows_per_block = 64;  // 4 waves x 16 rows
  hipLaunchKernelGGL(planar_flow_kernel, dim3(nrows / rows_per_block),
                     dim3(128), 0, stream, z, ws, zf, ldj);
}